// Gate_48825188221348
// MI455X (gfx1250) — compile-verified
//
#include <hip/hip_runtime.h>
#include <hip/hip_bf16.h>
#include <math.h>

#define DIM            2048
#define NEXPERTS       64
#define TOPK           8
#define NTOK           16384
#define WAVES_PER_BLK  8
#define ROWS_PER_WAVE  16
#define TOK_PER_BLK    (WAVES_PER_BLK * ROWS_PER_WAVE)   // 128
#define LDS_STRIDE     65                                 // 64 + pad (bank-conflict free)

typedef float v2f __attribute__((ext_vector_type(2)));
typedef float v8f __attribute__((ext_vector_type(8)));

__global__ __launch_bounds__(256) void Gate_48825188221348_kernel(
    const float* __restrict__ x,      // (NTOK, DIM)
    const float* __restrict__ w,      // (NEXPERTS, DIM)
    const float* __restrict__ bias,   // (NEXPERTS)
    float* __restrict__ out_vals,     // (NTOK, TOPK)
    float* __restrict__ out_idx,      // (NTOK, TOPK)  indices as float
    float* __restrict__ out_gate)     // (NTOK, NEXPERTS)
{
    __shared__ float              slog[TOK_PER_BLK * LDS_STRIDE];
    __shared__ float              ssum[TOK_PER_BLK];
    __shared__ unsigned long long smask[TOK_PER_BLK];
    __shared__ float              sbias[NEXPERTS];

    const int tid  = threadIdx.x;
    const int wave = tid >> 5;
    const int lane = tid & 31;
    const int half = lane >> 4;   // 0: lanes 0-15, 1: lanes 16-31
    const int l15  = lane & 15;

    if (tid < NEXPERTS) sbias[tid] = bias[tid];

    // ---------------- Phase 1: WMMA f32 GEMM, 16 tokens x 64 experts per wave
    const int tok_base = blockIdx.x * TOK_PER_BLK + wave * ROWS_PER_WAVE;

    // A fragment base: row = tok_base + l15, K lane-pair offset 2*half
    const float* aptr  = x + (size_t)(tok_base + l15) * DIM + 2 * half;
    // B fragment bases: expert = t*16 + l15, K lane-pair offset 2*half
    const float* bptr0 = w + (size_t)(l15 +  0) * DIM + 2 * half;
    const float* bptr1 = w + (size_t)(l15 + 16) * DIM + 2 * half;
    const float* bptr2 = w + (size_t)(l15 + 32) * DIM + 2 * half;
    const float* bptr3 = w + (size_t)(l15 + 48) * DIM + 2 * half;

    v8f acc0 = {}, acc1 = {}, acc2 = {}, acc3 = {};

#pragma unroll 4
    for (int k = 0; k < DIM; k += 4) {
        v2f a  = *(const v2f*)(aptr  + k);
        v2f b0 = *(const v2f*)(bptr0 + k);
        v2f b1 = *(const v2f*)(bptr1 + k);
        v2f b2 = *(const v2f*)(bptr2 + k);
        v2f b3 = *(const v2f*)(bptr3 + k);
        acc0 = __builtin_amdgcn_wmma_f32_16x16x4_f32(false, a, false, b0, (short)0, acc0, false, false);
        acc1 = __builtin_amdgcn_wmma_f32_16x16x4_f32(false, a, false, b1, (short)0, acc1, false, false);
        acc2 = __builtin_amdgcn_wmma_f32_16x16x4_f32(false, a, false, b2, (short)0, acc2, false, false);
        acc3 = __builtin_amdgcn_wmma_f32_16x16x4_f32(false, a, false, b3, (short)0, acc3, false, false);
    }

    // C/D layout: VGPR r, lanes 0-15 -> M=r, lanes 16-31 -> M=r+8; N = l15
#pragma unroll
    for (int r = 0; r < 8; ++r) {
        const int row = wave * ROWS_PER_WAVE + r + 8 * half;
        float* dst = &slog[row * LDS_STRIDE];
        dst[ 0 + l15] = acc0[r];
        dst[16 + l15] = acc1[r];
        dst[32 + l15] = acc2[r];
        dst[48 + l15] = acc3[r];
    }

    __syncthreads();

    // ---------------- Phase 2: per-token softmax + top-8 (threads 0..127)
    if (tid < TOK_PER_BLK) {
        float* l = &slog[tid * LDS_STRIDE];

        float m = -INFINITY;
#pragma unroll 8
        for (int e = 0; e < NEXPERTS; ++e)
            m = fmaxf(m, l[e] + sbias[e]);

        float s = 0.0f;
#pragma unroll 8
        for (int e = 0; e < NEXPERTS; ++e) {
            float p = __expf(l[e] + sbias[e] - m);   // unnormalized prob
            l[e] = p;
            s += p;
        }
        ssum[tid] = s;
        const float inv = 1.0f / s;

        const int gt = blockIdx.x * TOK_PER_BLK + tid;
        unsigned long long mask = 0ull;
#pragma unroll
        for (int j = 0; j < TOPK; ++j) {
            float best = -1.0f;
            int   bi   = 0;
            for (int e = 0; e < NEXPERTS; ++e) {
                if (!((mask >> e) & 1ull)) {
                    float p = l[e];
                    if (p > best) { best = p; bi = e; }   // ties -> lowest index
                }
            }
            mask |= (1ull << bi);
            out_vals[(size_t)gt * TOPK + j] = best * inv;
            out_idx [(size_t)gt * TOPK + j] = (float)bi;
        }
        smask[tid] = mask;
    }

    __syncthreads();

    // ---------------- Phase 3: cooperative coalesced dense-gate write
    const int base_gt = blockIdx.x * TOK_PER_BLK;
    for (int i = tid; i < TOK_PER_BLK * NEXPERTS; i += 256) {
        const int t = i >> 6;
        const int e = i & 63;
        const unsigned long long mk = smask[t];
        float v = ((mk >> e) & 1ull) ? slog[t * LDS_STRIDE + e] / ssum[t] : 0.0f;
        out_gate[(size_t)(base_gt + t) * NEXPERTS + e] = v;
    }
}

extern "C" void kernel_launch(void* const* d_in, const int* in_sizes, int n_in,
                              void* d_out, int out_size, void* d_ws, size_t ws_size,
                              hipStream_t stream) {
    const float* x    = (const float*)d_in[0];   // (16384, 2048)
    const float* w    = (const float*)d_in[1];   // (64, 2048)
    const float* bias = (const float*)d_in[2];   // (64,)

    float* out       = (float*)d_out;
    float* out_vals  = out;                          // 16384*8
    float* out_idx   = out + (size_t)NTOK * TOPK;    // 16384*8
    float* out_gate  = out + (size_t)NTOK * TOPK * 2;// 16384*64

    dim3 grid(NTOK / TOK_PER_BLK);   // 128 blocks
    dim3 block(256);                  // 8 waves (wave32)
    Gate_48825188221348_kernel<<<grid, block, 0, stream>>>(
        x, w, bias, out_vals, out_idx, out_gate);
}